// RadiusInteractionGraph_86689619902626
// MI455X (gfx1250) — compile-verified
//
#include <hip/hip_runtime.h>

// ---------------------------------------------------------------------------
// radius_graph(pos, r=10, batch, loop=False, max_num_neighbors=32) + lengths
// MI455X / gfx1250: wave32, WMMA. Distance tiles via V_WMMA_F32_16X16X4_F32
// (16x4 @ 4x16 -> 16x16 Gram tile; K=4 with zero-padded 4th coordinate).
// ---------------------------------------------------------------------------

typedef float v2f __attribute__((ext_vector_type(2)));
typedef float v8f __attribute__((ext_vector_type(8)));

#define KNBR   32
#define R2     100.0f
#define BIGF   3.402823e38f

// ---- Kernel 1: molecule segment bounds from sorted batch -------------------
__global__ void seg_bounds_kernel(const int* __restrict__ batch,
                                  int* __restrict__ segStart,
                                  int* __restrict__ segEnd, int N) {
    int i = blockIdx.x * blockDim.x + threadIdx.x;
    if (i >= N) return;
    int b = batch[i];
    if (i == 0     || batch[i - 1] != b) segStart[b] = i;
    if (i == N - 1 || batch[i + 1] != b) segEnd[b]   = i + 1;
}

// ---- Kernel 2: WMMA distance tiles + cutoff filter -> compacted candidates -
// One wave per 16-target tile. For each candidate tile of 16 atoms in the
// molecule segment: D = A x B (rank-3 Gram) with v_wmma_f32_16x16x4_f32,
// d2 = |t|^2 + |c|^2 - 2*dot, filter, append to per-row list (LDS counter;
// rows are wave-exclusive; DS ops are in-order within a wave -> no barrier).
__global__ void build_cand_kernel(const float* __restrict__ pos,
                                  const int*   __restrict__ batch,
                                  const int*   __restrict__ segStart,
                                  const int*   __restrict__ segEnd,
                                  float* __restrict__ candD2,
                                  int*   __restrict__ candIdx,
                                  int*   __restrict__ cnt,
                                  int N, int CAP) {
    __shared__ int lcnt[8][16];                    // 8 waves per 256-thr block
    const int lane = threadIdx.x & 31;
    const int wv   = threadIdx.x >> 5;
    const int tile = blockIdx.x * 8 + wv;
    const int rowBase = tile * 16;
    if (rowBase >= N) return;
    if (lane < 16) lcnt[wv][lane] = 0;

    // --- target-side data: lane holds row m = lane%16 (ISA A-matrix layout:
    //     lanes 0-15 -> K0,K1 ; lanes 16-31 -> K2,K3) -----------------------
    const int   m   = lane & 15;
    const int   im  = min(rowBase + m, N - 1);
    const float tx  = pos[im * 3 + 0];
    const float ty  = pos[im * 3 + 1];
    const float tz  = pos[im * 3 + 2];
    const float tsq = tx * tx + ty * ty + tz * tz;
    const int   tb  = batch[im];
    const v2f   A   = { (lane < 16) ? tx : tz,     // K0 / K2
                        (lane < 16) ? ty : 0.0f }; // K1 / K3 (zero pad)

    // Hoist loop-invariant row broadcasts out of the candidate loop:
    // element (M, n) with M = v + 8*(lane>=16) needs target row M's norm/batch.
    float tsqr[8]; int tbr[8];
    #pragma unroll
    for (int v = 0; v < 8; ++v) {
        const int M = v + ((lane >> 4) << 3);
        tsqr[v] = __shfl(tsq, M, 32);
        tbr[v]  = __shfl(tb,  M, 32);
    }

    const int segLo = segStart[batch[rowBase]];
    const int segHi = segEnd[batch[min(rowBase + 15, N - 1)]];

    for (int c = segLo; c < segHi; c += 16) {
        const bool jValid = (c + m) < segHi;
        const int  j      = min(c + m, segHi - 1);
        const float cx  = pos[j * 3 + 0];
        const float cy  = pos[j * 3 + 1];
        const float cz  = pos[j * 3 + 2];
        const float csq = cx * cx + cy * cy + cz * cz;
        const int   cb  = batch[j];
        // B 4x16 row-striped: VGPR0 = rows K0|K1, VGPR1 = rows K2|K3(=0)
        const v2f Bm = { (lane < 16) ? cx : cy,
                         (lane < 16) ? cz : 0.0f };

        v8f acc = {};
        acc = __builtin_amdgcn_wmma_f32_16x16x4_f32(
                  /*neg_a=*/false, A, /*neg_b=*/false, Bm,
                  /*c_mod=*/(short)0, acc, /*reuse_a=*/false, /*reuse_b=*/false);

        // D layout: column n = lane%16 (== this lane's candidate j),
        //           row M = v + 8*(lane>=16)
        #pragma unroll
        for (int v = 0; v < 8; ++v) {
            const int   M    = v + ((lane >> 4) << 3);
            const int   iM   = rowBase + M;
            const float d2   = tsqr[v] + csq - 2.0f * acc[v];
            const bool valid = jValid && (iM < N) && (cb == tbr[v]) &&
                               (j != iM) && (d2 <= R2);
            if (valid) {
                int p = atomicAdd(&lcnt[wv][M], 1);
                if (p < CAP) {
                    candD2 [iM * CAP + p] = d2;
                    candIdx[iM * CAP + p] = j;
                }
            }
        }
    }
    if (lane < 16 && rowBase + lane < N)
        cnt[rowBase + lane] = min(lcnt[wv][lane], CAP);
}

// ---- Kernel 3: exact top-K (K = 32 = wave32: one slot per lane) ------------
// Invariant: slots (sd2,sidx) sorted ascending across lanes. Per 32-candidate
// chunk: bitonic-sort chunk (15 stages), reverse, elementwise min vs slots
// (keeps 32 smallest of 64 -> bitonic), 5-stage merge cleanup. Final slots are
// already ascending (= top_k nearest-first order).
__global__ void topk_select_kernel(const float* __restrict__ candD2,
                                   const int*   __restrict__ candIdx,
                                   const int*   __restrict__ cnt,
                                   float* __restrict__ out,
                                   int N, int CAP) {
    const int lane = threadIdx.x & 31;
    const int wv   = threadIdx.x >> 5;
    const int i    = blockIdx.x * 8 + wv;
    if (i >= N) return;

    const int mcnt = min(cnt[i], CAP);
    float sd2  = BIGF;
    int   sidx = i;                                // pad = self-loop

    for (int base = 0; base < mcnt; base += 32) {
        float cd2 = BIGF; int cidx = i;
        if (base + lane < mcnt) {
            cd2  = candD2 [i * CAP + base + lane];
            cidx = candIdx[i * CAP + base + lane];
        }
        // full bitonic sort of the chunk, ascending
        #pragma unroll
        for (int k = 2; k <= 32; k <<= 1) {
            #pragma unroll
            for (int j = k >> 1; j > 0; j >>= 1) {
                float od2  = __shfl_xor(cd2,  j, 32);
                int   oidx = __shfl_xor(cidx, j, 32);
                const bool up        = ((lane & k) == 0);
                const bool lower     = ((lane & j) == 0);
                const bool takeOther = lower ? ((cd2 > od2) == up)
                                             : ((cd2 < od2) == up);
                if (takeOther) { cd2 = od2; cidx = oidx; }
            }
        }
        // reverse chunk (descending) and take elementwise min against slots:
        // result = 32 smallest of 64, bitonic sequence
        const float rd2  = __shfl(cd2,  31 - lane, 32);
        const int   ridx = __shfl(cidx, 31 - lane, 32);
        if (rd2 < sd2) { sd2 = rd2; sidx = ridx; }
        // bitonic merge cleanup -> ascending
        #pragma unroll
        for (int j = 16; j > 0; j >>= 1) {
            float od2  = __shfl_xor(sd2,  j, 32);
            int   oidx = __shfl_xor(sidx, j, 32);
            const bool lower     = ((lane & j) == 0);
            const bool takeOther = lower ? (sd2 > od2) : (sd2 < od2);
            if (takeOther) { sd2 = od2; sidx = oidx; }
        }
    }

    const bool valid = sd2 < 3.0e38f;
    const int  NK = N * KNBR;
    const int  e  = i * KNBR + lane;
    out[e]          = (float)sidx;                         // src (row)
    out[NK + e]     = (float)i;                            // tgt (col)
    out[2 * NK + e] = valid ? sqrtf(fmaxf(sd2, 0.0f)) : 0.0f;
}

// ---------------------------------------------------------------------------
extern "C" void kernel_launch(void* const* d_in, const int* in_sizes, int n_in,
                              void* d_out, int out_size, void* d_ws, size_t ws_size,
                              hipStream_t stream) {
    const float* pos   = (const float*)d_in[0];
    const int*   batch = (const int*)d_in[1];
    const int    N     = in_sizes[1];
    const int    NB    = 64;                       // molecule slots (B=32)

    // workspace: segStart[NB] | segEnd[NB] | cnt[N] | candD2[N*CAP] | candIdx[N*CAP]
    int* segStart = (int*)d_ws;
    int* segEnd   = segStart + NB;
    int* cnt      = segEnd + NB;
    const size_t fixedBytes = (size_t)(2 * NB + N) * sizeof(int);
    int CAP = 96;                                  // expected ~33 in-cutoff/row
    if (ws_size > fixedBytes) {
        size_t c = (ws_size - fixedBytes) / ((size_t)N * 8);
        if ((size_t)CAP > c) CAP = (int)c;
    }
    if (CAP < 1) CAP = 1;
    float* candD2  = (float*)(cnt + N);
    int*   candIdx = (int*)(candD2 + (size_t)N * CAP);

    seg_bounds_kernel<<<(N + 255) / 256, 256, 0, stream>>>(batch, segStart, segEnd, N);

    const int tiles = (N + 15) / 16;               // 16 targets per wave
    build_cand_kernel<<<(tiles + 7) / 8, 256, 0, stream>>>(
        pos, batch, segStart, segEnd, candD2, candIdx, cnt, N, CAP);

    topk_select_kernel<<<(N + 7) / 8, 256, 0, stream>>>(
        candD2, candIdx, cnt, (float*)d_out, N, CAP);
}